// RigidFieldLoss_54400055771862
// MI455X (gfx1250) — compile-verified
//
#include <hip/hip_runtime.h>

#define HH 96
#define WW 192
#define DD 192
#define VV (HH*WW*DD)
#define NCH 5

// workspace float layout
#define WS_C1    0      // 16x16 accumulator grid (masked coordinate moments)
#define WS_C2    256    // 16x16 accumulator grid (mask / s-mask moments)
#define WS_TP    512    // 16 rows x 4 : [R-I | t] per (channel,row), row 15 = 0
#define WS_FLAG  576    // 5 validity flags
#define WS_DENOM 584
#define WS_LOSS  585
#define WS_ZERO_N 1024

typedef float v2f __attribute__((ext_vector_type(2)));
typedef float v8f __attribute__((ext_vector_type(8)));

__device__ __forceinline__ float pick3(int comp, float x, float y, float z) {
  return (comp == 0) ? x : ((comp == 1) ? y : z);
}
__device__ __forceinline__ float det3(const float A[3][3]) {
  return A[0][0] * (A[1][1] * A[2][2] - A[1][2] * A[2][1])
       - A[0][1] * (A[1][0] * A[2][2] - A[1][2] * A[2][0])
       + A[0][2] * (A[1][0] * A[2][1] - A[1][1] * A[2][0]);
}

__global__ void k_init(float* ws) {
  int i = blockIdx.x * blockDim.x + threadIdx.x;
  if (i < WS_ZERO_N) ws[i] = 0.0f;
}

// ---------------------------------------------------------------------------
// Pass A: per-channel moments via WMMA f32 16x16x4, K = 4 voxels / instruction.
// c1 rows (M = 3n+i, n<5): A = y_n(v) * p_i(v)
// c2 rows: M<5: y_n(v); 5<=M<10: s_{M-5}(v)
// B cols:  c<3: p_c+f_c ; c==3: 1 ; 4<=c<7: p_{c-4} ; else 0
// Contiguous chunk per wave; coordinates maintained incrementally (step 4
// divides DD, so wraps are exact; half-wave +2 offset preserved via d -= DD).
// ---------------------------------------------------------------------------
__global__ void k_moments(const float* __restrict__ yoh,
                          const float* __restrict__ soh,
                          const float* __restrict__ flow,
                          float* __restrict__ ws) {
  const int lane = threadIdx.x & 31;
  const int Mrow = lane & 15;
  const int half = lane >> 4;
  const int wid  = blockIdx.x * (blockDim.x >> 5) + (threadIdx.x >> 5);
  const int nw   = gridDim.x * (blockDim.x >> 5);
  const float* yb = yoh + VV;   // skip background class 0
  const float* sb = soh + VV;

  // ---- lane-constant operand selectors (hoisted out of the loop) ----
  // B feature column c = Mrow
  const float bConst      = (Mrow == 3) ? 1.f : 0.f;
  const int   bCompRaw    = (Mrow < 3) ? Mrow : ((Mrow >= 4 && Mrow < 7) ? (Mrow - 4) : -1);
  const float bCoordScale = (bCompRaw >= 0) ? 1.f : 0.f;
  const int   bComp       = (bCompRaw >= 0) ? bCompRaw : 0;
  const float bFlowScale  = (Mrow < 3) ? 1.f : 0.f;
  const float* fplane     = flow + ((Mrow < 3) ? Mrow : 0) * VV;
  // A1 row = 3n+i : y_n * p_i
  const float a1Scale  = (Mrow < 15) ? 1.f : 0.f;
  const int   a1n      = (Mrow < 15) ? (Mrow / 3) : 0;
  const int   a1i      = (Mrow < 15) ? (Mrow - 3 * a1n) : 0;
  const float* a1plane = yb + a1n * VV;
  // A2 row: y_n (n<5) or s_n (5..9)
  const float a2Scale  = (Mrow < 2 * NCH) ? 1.f : 0.f;
  const float* a2plane = (Mrow < NCH) ? (yb + Mrow * VV)
                       : ((Mrow < 2 * NCH) ? (sb + (Mrow - NCH) * VV) : yb);

  // ---- contiguous chunk of 4-voxel groups for this wave ----
  const int ngrp  = VV / 4;
  const int chunk = ngrp / nw;
  const int rem   = ngrp - chunk * nw;
  const int start = wid * chunk + ((wid < rem) ? wid : rem);
  const int cnt   = chunk + ((wid < rem) ? 1 : 0);

  int v0 = start * 4 + 2 * half;          // this half-wave owns K = {2h, 2h+1}
  int h  = v0 / (WW * DD);
  int r  = v0 - h * (WW * DD);
  int w  = r / DD;
  int d  = r - w * DD;

  v8f c1 = {};
  v8f c2 = {};
  for (int it = 0; it < cnt; ++it) {
    const int v1 = v0 + 1;                // d+1 never wraps (d is even)
    const float x  = (float)h;
    const float y  = (float)w;
    const float z0 = (float)d;
    const float z1 = z0 + 1.f;

    const float b0 = bConst + bCoordScale * pick3(bComp, x, y, z0) + bFlowScale * fplane[v0];
    const float b1 = bConst + bCoordScale * pick3(bComp, x, y, z1) + bFlowScale * fplane[v1];
    const float a0 = a1Scale * a1plane[v0] * pick3(a1i, x, y, z0);
    const float a1 = a1Scale * a1plane[v1] * pick3(a1i, x, y, z1);
    const float e0 = a2Scale * a2plane[v0];
    const float e1 = a2Scale * a2plane[v1];

    v2f A1 = {a0, a1};
    v2f A2 = {e0, e1};
    v2f B  = {b0, b1};
    c1 = __builtin_amdgcn_wmma_f32_16x16x4_f32(false, A1, false, B, (short)0, c1, false, false);
    c2 = __builtin_amdgcn_wmma_f32_16x16x4_f32(false, A2, false, B, (short)0, c2, false, false);

    v0 += 4;
    d  += 4;
    if (d >= DD) {            // lane-uniform wrap (4 | DD)
      d -= DD;
      if (++w >= WW) { w -= WW; ++h; }
    }
  }

  // block-level reduction of the 2x(16x16) accumulator grids, then global atomics
  __shared__ float lds[512];
  for (int i = threadIdx.x; i < 512; i += blockDim.x) lds[i] = 0.f;
  __syncthreads();
  for (int k = 0; k < 8; ++k) {
    int row = half ? (k + 8) : k;         // C/D layout: VGPR k, upper lanes = M+8
    atomicAdd(&lds[row * 16 + Mrow], c1[k]);
    atomicAdd(&lds[256 + row * 16 + Mrow], c2[k]);
  }
  __syncthreads();
  for (int i = threadIdx.x; i < 512; i += blockDim.x)
    atomicAdd(&ws[i], lds[i]);
}

// ---------------------------------------------------------------------------
// Pass B: five 3x3 Kabsch solves (Jacobi eigen of H^T H), one channel/thread.
// ---------------------------------------------------------------------------
__global__ void k_solve(float* ws) {
  const int n = threadIdx.x;
  __shared__ float sDen[NCH];
  if (n < NCH) {
    const float* C1 = ws + WS_C1;
    const float* C2 = ws + WS_C2;
    float Mn = C2[n * 16 + 3];
    float sS = C2[(NCH + n) * 16 + 3];
    float dv[3], spv[3], bv[3], Ar[3][3];
    for (int j = 0; j < 3; ++j) {
      dv[j]  = C2[n * 16 + j];
      spv[j] = C2[(NCH + n) * 16 + 4 + j];
    }
    for (int i = 0; i < 3; ++i) {
      bv[i] = C1[(3 * n + i) * 16 + 3];
      for (int j = 0; j < 3; ++j) Ar[i][j] = C1[(3 * n + i) * 16 + j];
    }
    bool valid = (Mn > 100.f) && (sS > 100.f);
    float flag = 0.f;
    float Tp[3][4] = {};
    if (valid) {
      float ycm[3], scm[3];
      for (int i = 0; i < 3; ++i) { ycm[i] = bv[i] / Mn; scm[i] = spv[i] / sS; }
      float Hm[3][3];
      for (int i = 0; i < 3; ++i)
        for (int j = 0; j < 3; ++j)
          Hm[i][j] = Ar[i][j] - ycm[i] * dv[j] - bv[i] * scm[j] + Mn * ycm[i] * scm[j];
      float S[3][3];
      for (int i = 0; i < 3; ++i)
        for (int j = 0; j < 3; ++j) {
          float acc = 0.f;
          for (int k = 0; k < 3; ++k) acc += Hm[k][i] * Hm[k][j];
          S[i][j] = acc;
        }
      float Vm[3][3] = {{1,0,0},{0,1,0},{0,0,1}};
      for (int sweep = 0; sweep < 12; ++sweep) {
        for (int pair = 0; pair < 3; ++pair) {
          int p = (pair == 2) ? 1 : 0;
          int q = (pair == 0) ? 1 : 2;
          float apq = S[p][q];
          if (fabsf(apq) > 1e-20f) {
            float tau = (S[q][q] - S[p][p]) / (2.f * apq);
            float t = ((tau >= 0.f) ? 1.f : -1.f) / (fabsf(tau) + sqrtf(1.f + tau * tau));
            float c = 1.f / sqrtf(1.f + t * t);
            float s = t * c;
            for (int k = 0; k < 3; ++k) { float kp=S[k][p], kq=S[k][q]; S[k][p]=c*kp-s*kq; S[k][q]=s*kp+c*kq; }
            for (int k = 0; k < 3; ++k) { float pk=S[p][k], qk=S[q][k]; S[p][k]=c*pk-s*qk; S[q][k]=s*pk+c*qk; }
            for (int k = 0; k < 3; ++k) { float kp=Vm[k][p], kq=Vm[k][q]; Vm[k][p]=c*kp-s*kq; Vm[k][q]=s*kp+c*kq; }
          }
        }
      }
      float lam[3] = {S[0][0], S[1][1], S[2][2]};
      for (int pass = 0; pass < 3; ++pass) {
        int a = (pass == 1) ? 1 : 0, b = a + 1;
        if (lam[a] < lam[b]) {
          float tl = lam[a]; lam[a] = lam[b]; lam[b] = tl;
          for (int k = 0; k < 3; ++k) { float tv = Vm[k][a]; Vm[k][a] = Vm[k][b]; Vm[k][b] = tv; }
        }
      }
      float Um[3][3];
      for (int k = 0; k < 3; ++k) {
        float u[3];
        for (int i = 0; i < 3; ++i) {
          float acc = 0.f;
          for (int j = 0; j < 3; ++j) acc += Hm[i][j] * Vm[j][k];
          u[i] = acc;
        }
        float nrm = sqrtf(u[0]*u[0] + u[1]*u[1] + u[2]*u[2]);
        if (nrm > 1e-6f) {
          for (int i = 0; i < 3; ++i) Um[i][k] = u[i] / nrm;
        } else if (k == 2) {
          Um[0][2] = Um[1][0]*Um[2][1] - Um[2][0]*Um[1][1];
          Um[1][2] = Um[2][0]*Um[0][1] - Um[0][0]*Um[2][1];
          Um[2][2] = Um[0][0]*Um[1][1] - Um[1][0]*Um[0][1];
        } else {
          for (int i = 0; i < 3; ++i) Um[i][k] = (i == k) ? 1.f : 0.f;
        }
      }
      float dsign = (det3(Um) * det3(Vm) >= 0.f) ? 1.f : -1.f;
      float R[3][3];
      for (int i = 0; i < 3; ++i)
        for (int j = 0; j < 3; ++j)
          R[i][j] = Vm[i][0]*Um[j][0] + Vm[i][1]*Um[j][1] + dsign*Vm[i][2]*Um[j][2];
      for (int p = 0; p < 3; ++p) {
        float acc = scm[p];
        for (int q = 0; q < 3; ++q) acc -= R[p][q] * ycm[q];
        for (int q = 0; q < 3; ++q) Tp[p][q] = R[p][q] - ((p == q) ? 1.f : 0.f);
        Tp[p][3] = acc;
      }
      flag = 1.f;
    }
    for (int p = 0; p < 3; ++p)
      for (int q = 0; q < 4; ++q)
        ws[WS_TP + (3 * n + p) * 4 + q] = Tp[p][q];
    ws[WS_FLAG + n] = flag;
    sDen[n] = flag * Mn;
  }
  __syncthreads();
  if (threadIdx.x == 0) {
    float den = 0.f;
    for (int k = 0; k < NCH; ++k) den += sDen[k];
    ws[WS_DENOM] = den;
  }
}

// ---------------------------------------------------------------------------
// Pass C: apply [R-I|t] to 16 voxels per WMMA (A = constant 15x4 transform
// block, B = homogeneous coords built in ALU with incremental (h,w,d)),
// mask, combine across lane pairs with shfl_xor(16), vector-norm reduce.
// ---------------------------------------------------------------------------
__global__ void k_apply(const float* __restrict__ yoh,
                        const float* __restrict__ flow,
                        float* __restrict__ ws) {
  const int lane = threadIdx.x & 31;
  const int col  = lane & 15;
  const int half = lane >> 4;
  const int wid  = blockIdx.x * (blockDim.x >> 5) + (threadIdx.x >> 5);
  const int nw   = gridDim.x * (blockDim.x >> 5);
  const float* yb = yoh + VV;

  v2f A;
  A[0] = ws[WS_TP + col * 4 + 2 * half];
  A[1] = ws[WS_TP + col * 4 + 2 * half + 1];
  const float fl0 = ws[WS_FLAG + 0], fl1 = ws[WS_FLAG + 1], fl2 = ws[WS_FLAG + 2];
  const float fl3 = ws[WS_FLAG + 3], fl4 = ws[WS_FLAG + 4];
  // lane-constant channel plane pointers / flags for the masked combine
  const float* pm0 = yb + (half ? 2 : 0) * VV;
  const float* pm1 = yb + (half ? 3 : 1) * VV;
  const float* pm2 = yb + (half ? 4 : 2) * VV;
  const float sc0 = half ? fl2 : fl0;
  const float sc1 = half ? fl3 : fl1;
  const float sc2 = half ? fl4 : fl2;

  const int ngrp  = VV / 16;
  const int chunk = ngrp / nw;
  const int rem   = ngrp - chunk * nw;
  const int start = wid * chunk + ((wid < rem) ? wid : rem);
  const int cnt   = chunk + ((wid < rem) ? 1 : 0);

  int vbase = start * 16;
  int h = vbase / (WW * DD);
  int r = vbase - h * (WW * DD);
  int w = r / DD;
  int dbase = r - w * DD;          // 16 | DD, so col stays within the row

  float local = 0.f;
  for (int it = 0; it < cnt; ++it) {
    const int vox = vbase + col;
    v2f B;
    if (half == 0) { B[0] = (float)h;            B[1] = (float)w; }   // K=0,1
    else           { B[0] = (float)(dbase + col); B[1] = 1.f; }       // K=2,3
    v8f dacc = {};
    dacc = __builtin_amdgcn_wmma_f32_16x16x4_f32(false, A, false, B, (short)0, dacc, false, false);

    const float m0 = pm0[vox] * sc0;
    const float m1 = pm1[vox] * sc1;
    const float m2 = pm2[vox] * sc2;
    float pr0, pr1, pr2, pS;
    if (half == 0) {   // lane owns rows 0..7 of voxel `col`
      pr0 = m0 * dacc[0] + m1 * dacc[3] + m2 * dacc[6];
      pr1 = m0 * dacc[1] + m1 * dacc[4] + m2 * dacc[7];
      pr2 = m0 * dacc[2] + m1 * dacc[5];
      pS  = m0 + m1;                 // ch2 counted by the upper half
    } else {           // lane owns rows 8..14
      pr0 = m1 * dacc[1] + m2 * dacc[4];
      pr1 = m1 * dacc[2] + m2 * dacc[5];
      pr2 = m0 * dacc[0] + m1 * dacc[3] + m2 * dacc[6];
      pS  = m0 + m1 + m2;
    }
    const float tr0 = pr0 + __shfl_xor(pr0, 16, 32);
    const float tr1 = pr1 + __shfl_xor(pr1, 16, 32);
    const float tr2 = pr2 + __shfl_xor(pr2, 16, 32);
    const float tS  = pS  + __shfl_xor(pS,  16, 32);

    const float dx = tr0 - tS * flow[0 * VV + vox];
    const float dy = tr1 - tS * flow[1 * VV + vox];
    const float dz = tr2 - tS * flow[2 * VV + vox];
    const float sq = dx * dx + dy * dy + dz * dz;
    const float contrib = (sq > 0.f) ? sqrtf(sq) : 0.f;
    if (half == 0) local += contrib;   // count each voxel once

    vbase += 16;
    dbase += 16;
    if (dbase >= DD) {                 // lane-uniform wrap (16 | DD)
      dbase -= DD;
      if (++w >= WW) { w -= WW; ++h; }
    }
  }

  __shared__ float red[256];
  red[threadIdx.x] = local;
  __syncthreads();
  for (int s = blockDim.x >> 1; s > 0; s >>= 1) {
    if (threadIdx.x < s) red[threadIdx.x] += red[threadIdx.x + s];
    __syncthreads();
  }
  if (threadIdx.x == 0) atomicAdd(&ws[WS_LOSS], red[0]);
}

__global__ void k_final(const float* __restrict__ ws, float* __restrict__ out) {
  float den = ws[WS_DENOM];
  out[0] = (den > 0.f) ? (ws[WS_LOSS] / den) : 0.f;
}

extern "C" void kernel_launch(void* const* d_in, const int* in_sizes, int n_in,
                              void* d_out, int out_size, void* d_ws, size_t ws_size,
                              hipStream_t stream) {
  (void)in_sizes; (void)n_in; (void)out_size; (void)ws_size;
  const float* yoh  = (const float*)d_in[0];
  const float* soh  = (const float*)d_in[1];
  const float* flow = (const float*)d_in[2];
  float* ws  = (float*)d_ws;
  float* out = (float*)d_out;

  k_init   <<<4,   256, 0, stream>>>(ws);
  k_moments<<<512, 256, 0, stream>>>(yoh, soh, flow, ws);
  k_solve  <<<1,   32,  0, stream>>>(ws);
  k_apply  <<<512, 256, 0, stream>>>(yoh, flow, ws);
  k_final  <<<1,   1,   0, stream>>>(ws, out);
}